// HFNO_2D_4741643895638
// MI455X (gfx1250) — compile-verified
//
#include <hip/hip_runtime.h>
#include <math.h>

// ---------------- problem constants ----------------
#define RES_     64
#define DEPTH_   32
#define INCH_    3
#define OUTCH_   4
#define BATCH_   64
#define HID_     1024
#define KYN_     33                     // RES/2+1
#define PLANE_   (RES_*KYN_)            // 2112
#define NEGS_    0.01f
#define TWO_PI_  6.28318530717958647692f

typedef __attribute__((ext_vector_type(2))) float v2f;
typedef __attribute__((ext_vector_type(8))) float v8f;

// Paired activation layout ("wmma-ready" A operand), K even:
//   AP[(k>>1)*128 + m*2 + (k&1)]  ==  A[m][k]   (m = 0..63)
// -> lane m reads (A[m,k], A[m,k+1]) as one aligned 8-byte load.

// =====================================================================
// 1) projection: h[d,x,y,b] = sum_c x[b,x,y,c]*Pw[c,d] + Pb[d]
// =====================================================================
__global__ void proj_kernel(const float* __restrict__ xin,
                            const float* __restrict__ Pw,
                            const float* __restrict__ Pb,
                            float* __restrict__ h, int total)
{
    int idx = blockIdx.x * blockDim.x + threadIdx.x;
    if (idx >= total) return;
    int b  = idx & 63;
    int t  = idx >> 6;
    int y  = t & 63;  t >>= 6;
    int xx = t & 63;  t >>= 6;
    int d  = t;                                   // 0..31
    const float* xp = xin + ((size_t)((b * RES_ + xx) * RES_ + y)) * INCH_;
    float acc = Pb[d];
    acc += xp[0] * Pw[0 * DEPTH_ + d];
    acc += xp[1] * Pw[1 * DEPTH_ + d];
    acc += xp[2] * Pw[2 * DEPTH_ + d];
    h[idx] = acc;
}

// =====================================================================
// 2) forward DFT along y: G[d,x,ky,b] = sum_y h[d,x,y,b] e^{-2pi i ky y/64}
// =====================================================================
__global__ void dft_y_kernel(const float* __restrict__ h,
                             float* __restrict__ GRe, float* __restrict__ GIm,
                             int total)
{
    __shared__ float ct[64], st[64];
    if (threadIdx.x < 64) {
        float s, c;
        sincosf(TWO_PI_ * (float)threadIdx.x / 64.0f, &s, &c);
        ct[threadIdx.x] = c; st[threadIdx.x] = s;
    }
    __syncthreads();
    int idx = blockIdx.x * blockDim.x + threadIdx.x;
    if (idx >= total) return;
    int b  = idx & 63;
    int t  = idx >> 6;
    int ky = t % KYN_;  t /= KYN_;
    int xx = t & 63;
    int d  = t >> 6;
    const float* hp = h + ((size_t)((d * 64 + xx) * 64)) * 64 + b;
    float re = 0.f, im = 0.f;
    for (int y = 0; y < 64; ++y) {
        float v = hp[(size_t)y * 64];
        int a = (ky * y) & 63;
        re += v * ct[a];
        im -= v * st[a];
    }
    GRe[idx] = re; GIm[idx] = im;
}

// =====================================================================
// 3) forward DFT along x: FX[d,kx,ky,b] = sum_x G[d,x,ky,b] e^{-2pi i kx x/64}
// =====================================================================
__global__ void dft_x_kernel(const float* __restrict__ GRe, const float* __restrict__ GIm,
                             float* __restrict__ FXRe, float* __restrict__ FXIm,
                             int total)
{
    __shared__ float ct[64], st[64];
    if (threadIdx.x < 64) {
        float s, c;
        sincosf(TWO_PI_ * (float)threadIdx.x / 64.0f, &s, &c);
        ct[threadIdx.x] = c; st[threadIdx.x] = s;
    }
    __syncthreads();
    int idx = blockIdx.x * blockDim.x + threadIdx.x;
    if (idx >= total) return;
    int b  = idx & 63;
    int t  = idx >> 6;
    int ky = t % KYN_;  t /= KYN_;
    int kx = t & 63;
    int d  = t >> 6;
    const float* grp = GRe + ((size_t)(d * PLANE_ + ky)) * 64 + b;
    const float* gip = GIm + ((size_t)(d * PLANE_ + ky)) * 64 + b;
    float re = 0.f, im = 0.f;
    for (int x = 0; x < 64; ++x) {
        float gr = grp[(size_t)x * KYN_ * 64];
        float gi = gip[(size_t)x * KYN_ * 64];
        int a = (kx * x) & 63;
        float c = ct[a], s = st[a];
        re += gr * c + gi * s;          // (gr + i gi) * (c - i s)
        im += gi * c - gr * s;
    }
    FXRe[idx] = re; FXIm[idx] = im;
}

// =====================================================================
// 4) per-band (ix,iy) lists, np.nonzero (row-major) order; pos = ix*33+iy
// =====================================================================
__global__ void band_pos_kernel(int* __restrict__ pos0, int* __restrict__ pos1,
                                int* __restrict__ pos2)
{
    if (threadIdx.x != 0) return;
    int band = blockIdx.x;
    int* pos = (band == 0) ? pos0 : (band == 1 ? pos1 : pos2);
    int ke  = (band == 0) ? 8 : (band == 1 ? 16 : 32);
    int ks  = (band == 0) ? 0 : (band == 1 ? 8 : 16);
    int ke2 = ke * ke, ks2 = ks * ks;
    int cnt = 0;
    for (int ix = 0; ix < 64; ++ix) {
        int kx = (ix < 32) ? ix : ix - 64;
        for (int iy = 0; iy < KYN_; ++iy) {
            int ky = (iy < 32) ? iy : 32;            // column 32 is ky=-32
            int r2 = kx * kx + ky * ky;
            if (r2 >= ks2 && r2 <= ke2) pos[cnt++] = ix * KYN_ + iy;
        }
    }
}

// =====================================================================
// 5) gather into paired A layout: k = d*2n + part*n + m
//    dest idx: p = idx>>7 (K-pair), b = (idx>>1)&63, klo = idx&1
// =====================================================================
__global__ void gather_kernel(const float* __restrict__ FXRe, const float* __restrict__ FXIm,
                              const int* __restrict__ pos,
                              float* __restrict__ V0P, int n, int total)
{
    int idx = blockIdx.x * blockDim.x + threadIdx.x;
    if (idx >= total) return;
    int p   = idx >> 7;
    int b   = (idx >> 1) & 63;
    int k   = 2 * p + (idx & 1);
    int d   = k / (2 * n);
    int r   = k - d * 2 * n;
    int part = (r >= n) ? 1 : 0;
    int m   = r - part * n;
    size_t src = ((size_t)d * PLANE_ + pos[m]) * 64 + b;
    V0P[idx] = part ? FXIm[src] : FXRe[src];
}

// =====================================================================
// 6) split-K WMMA GEMM:  Y[64,N] += A[64,K] @ W[K,N]
//    A given in paired layout AP (see above); W row-major K x N.
//    block = 256 threads = 8 waves; wave tile 16x16 (V_WMMA_F32_16X16X4_F32)
//    block tile M=64 x N=32; grid = (ceil(N/32), S splits); atomic accum.
//    Body is shared; the N=1024 instantiation gets compile-time address
//    math (immediate load offsets) for the dominant weight-stream GEMMs.
// =====================================================================
__device__ __forceinline__
void gemm_body(const float* __restrict__ AP, const float* __restrict__ W,
               float* __restrict__ Y, int K, int N, int Kc)
{
    const int lane  = threadIdx.x & 31;
    const int wave  = threadIdx.x >> 5;     // 0..7
    const int mtile = wave & 3;             // M tile (0..3)
    const int ntile = wave >> 2;            // 0..1
    const int l16   = lane & 15;
    const int kh    = lane >> 4;            // 0/1 -> K pairs {0,1} vs {2,3}
    const int ncol  = blockIdx.x * 32 + ntile * 16 + l16;
    const bool nvalid = (ncol < N);
    const int ncl   = nvalid ? ncol : (N - 1);   // clamped: garbage only feeds
                                                 // this (unstored) output column
    int k0 = blockIdx.y * Kc;
    int k1 = k0 + Kc; if (k1 > K) k1 = K;

    v8f c = {0.f, 0.f, 0.f, 0.f, 0.f, 0.f, 0.f, 0.f};

    // A: one aligned 8B load per lane per step (lane-coalesced 256B/wave)
    const float* ap = AP + ((size_t)(k0 >> 1) + kh) * 128 + l16 * 2;
    // W: two b32 loads per step, rows k and k+1 (lane-coalesced 64B each)
    const float* wp = W + (size_t)(k0 + 2 * kh) * N + ncl;

    #pragma unroll 8
    for (int k = k0; k < k1; k += 4) {
        v2f a = *(const v2f*)ap;
        v2f b;
        b.x = __builtin_nontemporal_load(wp);        // streamed once: NT hint
        b.y = __builtin_nontemporal_load(wp + N);
        c = __builtin_amdgcn_wmma_f32_16x16x4_f32(
                /*neg_a=*/false, a, /*neg_b=*/false, b,
                /*c_mod=*/(short)0, c, /*reuse_a=*/false, /*reuse_b=*/false);
        ap += 256;                                   // 4 K-values = 2 pairs
        wp += (size_t)4 * N;
        __builtin_prefetch(wp + (size_t)16 * N, 0, 1);  // speculative: safe OOB
    }

    if (nvalid) {
        #pragma unroll
        for (int r = 0; r < 8; ++r) {
            int M = mtile * 16 + r + kh * 8;     // C/D layout: lanes16-31 -> M+8
            atomicAdd(&Y[(size_t)M * N + ncol], c[r]);
        }
    }
}

template<int NFIX>
__global__ void gemm_splitk_fixed(const float* __restrict__ AP, const float* __restrict__ W,
                                  float* __restrict__ Y, int K, int Kc)
{
    gemm_body(AP, W, Y, K, NFIX, Kc);
}

__global__ void gemm_splitk_dyn(const float* __restrict__ AP, const float* __restrict__ W,
                                float* __restrict__ Y, int K, int N, int Kc)
{
    gemm_body(AP, W, Y, K, N, Kc);
}

// =====================================================================
// 7) bias + (optional) leaky-relu; emit next layer's paired A operand.
//    dest idx: p = idx>>7, n = 2p + (idx&1), m = (idx>>1)&63
// =====================================================================
__global__ void bias_act_pair_kernel(const float* __restrict__ Y, const float* __restrict__ bias,
                                     float* __restrict__ ZP, int N, int act, int total)
{
    int idx = blockIdx.x * blockDim.x + threadIdx.x;
    if (idx >= total) return;
    int p = idx >> 7;
    int m = (idx >> 1) & 63;
    int n = 2 * p + (idx & 1);
    float v = Y[(size_t)m * N + n] + bias[n];
    if (act) v = (v > 0.f) ? v : NEGS_ * v;
    ZP[idx] = v;
}

// =====================================================================
// 8) scatter band output into spectrum (o, plane, b). VP is paired, and
//    (re,im) for column pair (o*2n+2m, +1) sit adjacent -> one v2f load.
// =====================================================================
__global__ void scatter_kernel(const float* __restrict__ VP,   // paired (8n) x 64
                               const int* __restrict__ pos,
                               float* __restrict__ specRe, float* __restrict__ specIm,
                               int n, int total)
{
    int idx = blockIdx.x * blockDim.x + threadIdx.x;
    if (idx >= total) return;
    int b = idx & 63;
    int t = idx >> 6;
    int m = t % n;
    int o = t / n;
    v2f v = *(const v2f*)(VP + ((size_t)(o * n + m)) * 128 + b * 2);
    size_t di = ((size_t)o * PLANE_ + pos[m]) * 64 + b;
    specRe[di] += v.x;
    specIm[di] += v.y;
}

// =====================================================================
// 9) inverse DFT along x: g[o,x,ky,b] = sum_kx spec[o,kx,ky,b] e^{+2pi i kx x/64}
// =====================================================================
__global__ void idft_x_kernel(const float* __restrict__ specRe, const float* __restrict__ specIm,
                              float* __restrict__ gRe, float* __restrict__ gIm, int total)
{
    __shared__ float ct[64], st[64];
    if (threadIdx.x < 64) {
        float s, c;
        sincosf(TWO_PI_ * (float)threadIdx.x / 64.0f, &s, &c);
        ct[threadIdx.x] = c; st[threadIdx.x] = s;
    }
    __syncthreads();
    int idx = blockIdx.x * blockDim.x + threadIdx.x;
    if (idx >= total) return;
    int b  = idx & 63;
    int t  = idx >> 6;
    int ky = t % KYN_;  t /= KYN_;
    int xx = t & 63;
    int o  = t >> 6;
    const float* srp = specRe + ((size_t)o * PLANE_ + ky) * 64 + b;
    const float* sip = specIm + ((size_t)o * PLANE_ + ky) * 64 + b;
    float re = 0.f, im = 0.f;
    for (int kx = 0; kx < 64; ++kx) {
        float sr = srp[(size_t)kx * KYN_ * 64];
        float si = sip[(size_t)kx * KYN_ * 64];
        int a = (kx * xx) & 63;
        float c = ct[a], s = st[a];
        re += sr * c - si * s;          // (sr + i si) * (c + i s)
        im += sr * s + si * c;
    }
    gRe[idx] = re; gIm[idx] = im;
}

// =====================================================================
// 10) inverse real DFT along y + final transpose to (B,RES,RES,OUT)
// =====================================================================
__global__ void idft_y_out_kernel(const float* __restrict__ gRe, const float* __restrict__ gIm,
                                  float* __restrict__ out, int total)
{
    __shared__ float ct[64], st[64];
    if (threadIdx.x < 64) {
        float s, c;
        sincosf(TWO_PI_ * (float)threadIdx.x / 64.0f, &s, &c);
        ct[threadIdx.x] = c; st[threadIdx.x] = s;
    }
    __syncthreads();
    int idx = blockIdx.x * blockDim.x + threadIdx.x;
    if (idx >= total) return;
    int o  = idx & 3;
    int t  = idx >> 2;
    int y  = t & 63;  t >>= 6;
    int xx = t & 63;
    int b  = t >> 6;
    const float* grp = gRe + ((size_t)(o * 64 + xx) * KYN_) * 64 + b;
    const float* gip = gIm + ((size_t)(o * 64 + xx) * KYN_) * 64 + b;
    float acc = grp[0];                              // ky = 0
    float t32 = grp[(size_t)32 * 64];                // ky = 32, weight (-1)^y
    acc += (y & 1) ? -t32 : t32;
    for (int ky = 1; ky < 32; ++ky) {
        float gr = grp[(size_t)ky * 64];
        float gi = gip[(size_t)ky * 64];
        int a = (ky * y) & 63;
        acc += 2.f * (gr * ct[a] - gi * st[a]);
    }
    out[idx] = acc * (1.0f / 4096.0f);
}

// =====================================================================
// host side
// =====================================================================
static inline int ceil_div_i(int a, int b) { return (a + b - 1) / b; }

extern "C" void kernel_launch(void* const* d_in, const int* in_sizes, int n_in,
                              void* d_out, int out_size, void* d_ws, size_t ws_size,
                              hipStream_t stream)
{
    const float* xin = (const float*)d_in[0];
    const float* Pw  = (const float*)d_in[1];
    const float* Pb  = (const float*)d_in[2];
    const float* Wt[3][3];
    const float* Bt[3][3];
    int nband[3];
    for (int i = 0; i < 3; ++i) {
        for (int j = 0; j < 3; ++j) {
            Wt[i][j] = (const float*)d_in[3 + i * 6 + 2 * j];
            Bt[i][j] = (const float*)d_in[4 + i * 6 + 2 * j];
        }
        nband[i] = in_sizes[4 + i * 6 + 4] / (2 * OUTCH_);   // |b_i2| = 2*n*OUT
    }

    // ---- workspace layout (floats), recomputed identically every call ----
    float* ws = (float*)d_ws;
    size_t off = 0;
    const size_t HSZ  = (size_t)DEPTH_ * RES_ * RES_ * BATCH_;     // 8,388,608
    const size_t GSZ  = (size_t)DEPTH_ * PLANE_ * BATCH_;          // 4,325,376
    const size_t SSZ  = (size_t)OUTCH_ * PLANE_ * BATCH_;          //   540,672
    int maxN = 8 * nband[2]; if (maxN < HID_) maxN = HID_;

    float* h    = ws + off; off += HSZ;      // reused as V0P for band GEMMs
    float* GRe  = ws + off; off += GSZ;
    float* GIm  = ws + off; off += GSZ;
    float* FXRe = ws + off; off += GSZ;
    float* FXIm = ws + off; off += GSZ;
    float* Ybuf = ws + off; off += (size_t)BATCH_ * maxN;
    float* Z1   = ws + off; off += (size_t)HID_ * BATCH_;
    float* Z2   = ws + off; off += (size_t)HID_ * BATCH_;
    float* VP   = ws + off; off += (size_t)BATCH_ * maxN;
    float* sRe  = ws + off; off += SSZ;
    float* sIm  = ws + off; off += SSZ;
    float* gRe  = ws + off; off += SSZ;
    float* gIm  = ws + off; off += SSZ;
    int* pos0 = (int*)(ws + off); off += 2048;
    int* pos1 = (int*)(ws + off); off += 2048;
    int* pos2 = (int*)(ws + off); off += 2048;
    int* posB[3] = {pos0, pos1, pos2};

    const int TPB = 256;

    // 1) projection
    {
        int total = (int)HSZ;
        proj_kernel<<<ceil_div_i(total, TPB), TPB, 0, stream>>>(xin, Pw, Pb, h, total);
    }
    // 2-3) rfft2 as two DFT passes
    {
        int total = (int)GSZ;
        dft_y_kernel<<<ceil_div_i(total, TPB), TPB, 0, stream>>>(h, GRe, GIm, total);
        dft_x_kernel<<<ceil_div_i(total, TPB), TPB, 0, stream>>>(GRe, GIm, FXRe, FXIm, total);
    }
    // 4) band index lists
    band_pos_kernel<<<3, 32, 0, stream>>>(pos0, pos1, pos2);

    // spectrum accumulators
    hipMemsetAsync(sRe, 0, SSZ * sizeof(float), stream);
    hipMemsetAsync(sIm, 0, SSZ * sizeof(float), stream);

    // GEMM launcher: fixed-N path (immediate address math) for hidden width
    auto launch_gemm = [&](const float* AP, const float* W, float* Y, int K, int N) {
        hipMemsetAsync(Y, 0, (size_t)BATCH_ * N * sizeof(float), stream);
        int S = K / 2048; if (S < 1) S = 1; if (S > 32) S = 32;
        int Kc = ((K + S - 1) / S + 3) & ~3;
        dim3 grid(ceil_div_i(N, 32), S);
        if (N == HID_)
            gemm_splitk_fixed<HID_><<<grid, 256, 0, stream>>>(AP, W, Y, K, Kc);
        else
            gemm_splitk_dyn<<<grid, 256, 0, stream>>>(AP, W, Y, K, N, Kc);
    };

    // 5-8) per-band MLP pipelines (serialized on stream; band-overlap points
    //      handled by ordered += across scatter launches)
    for (int i = 0; i < 3; ++i) {
        int n  = nband[i];
        int K0 = 2 * DEPTH_ * n;        // 64n
        int N2 = 2 * n * OUTCH_;        // 8n
        float* V0P = h;                 // reuse projection buffer (K0*64 <= HSZ)

        {
            int total = K0 * BATCH_;
            gather_kernel<<<ceil_div_i(total, TPB), TPB, 0, stream>>>(
                FXRe, FXIm, posB[i], V0P, n, total);
        }
        // layer 0: (64 x K0) @ (K0 x 1024) -> leaky
        launch_gemm(V0P, Wt[i][0], Ybuf, K0, HID_);
        bias_act_pair_kernel<<<ceil_div_i(BATCH_ * HID_, TPB), TPB, 0, stream>>>(
            Ybuf, Bt[i][0], Z1, HID_, 1, BATCH_ * HID_);
        // layer 1: (64 x 1024) @ (1024 x 1024) -> leaky
        launch_gemm(Z1, Wt[i][1], Ybuf, HID_, HID_);
        bias_act_pair_kernel<<<ceil_div_i(BATCH_ * HID_, TPB), TPB, 0, stream>>>(
            Ybuf, Bt[i][1], Z2, HID_, 1, BATCH_ * HID_);
        // layer 2: (64 x 1024) @ (1024 x 8n), bias only
        launch_gemm(Z2, Wt[i][2], Ybuf, HID_, N2);
        bias_act_pair_kernel<<<ceil_div_i(BATCH_ * N2, TPB), TPB, 0, stream>>>(
            Ybuf, Bt[i][2], VP, N2, 0, BATCH_ * N2);
        // scatter into spectrum
        {
            int total = BATCH_ * OUTCH_ * n;
            scatter_kernel<<<ceil_div_i(total, TPB), TPB, 0, stream>>>(
                VP, posB[i], sRe, sIm, n, total);
        }
    }

    // 9-10) irfft2 + output transpose
    {
        int total = (int)SSZ;
        idft_x_kernel<<<ceil_div_i(total, TPB), TPB, 0, stream>>>(sRe, sIm, gRe, gIm, total);
    }
    {
        int total = BATCH_ * RES_ * RES_ * OUTCH_;   // == out_size
        idft_y_out_kernel<<<ceil_div_i(total, TPB), TPB, 0, stream>>>(
            gRe, gIm, (float*)d_out, total);
    }
    (void)n_in; (void)ws_size; (void)out_size;
}